// GAT_15453292331026
// MI455X (gfx1250) — compile-verified
//
#include <hip/hip_runtime.h>
#include <hip/hip_bf16.h>
#include <math.h>

#define N_NODES   100000
#define N_EDGES   1600000
#define E2        (N_EDGES + N_NODES)   // with self loops
#define F         128
#define NHID      64
#define NUM_GRAPHS 64

typedef float v2f __attribute__((ext_vector_type(2)));
typedef float v8f __attribute__((ext_vector_type(8)));

__device__ __forceinline__ float selu_f(float x) {
  const float scale = 1.0507009873554805f;
  const float alpha = 1.6732632423543772f;
  return x > 0.f ? scale * x : scale * alpha * (expf(x) - 1.f);
}

// order-preserving float<->uint map for atomic max
__device__ __forceinline__ unsigned enc_f(float f) {
  unsigned b = __float_as_uint(f);
  return (b & 0x80000000u) ? ~b : (b | 0x80000000u);
}
__device__ __forceinline__ float dec_f(unsigned u) {
  return __uint_as_float((u & 0x80000000u) ? (u ^ 0x80000000u) : ~u);
}

// -------- GEMM: C[N x 128] = A[N x 128] @ B[128 x 128], f32 WMMA 16x16x4 ----
// grid.x = N/16 (=6250, exact). 256 threads = 8 waves; wave w owns N-tile w.
__global__ __launch_bounds__(256) void gat_gemm_f32(
    const float* __restrict__ A, const float* __restrict__ B,
    float* __restrict__ C) {
  __shared__ float lds[16 * 132];           // pad 132 to avoid bank conflicts
  const int t  = threadIdx.x;
  const int m0 = blockIdx.x << 4;

  // stage 16x128 A tile (512 float4 by 256 threads)
  for (int i = t; i < 512; i += 256) {
    const int row = i >> 5, c4 = (i & 31) << 2;
    const float4 v = *(const float4*)(A + (size_t)(m0 + row) * F + c4);
    float* p = &lds[row * 132 + c4];
    p[0] = v.x; p[1] = v.y; p[2] = v.z; p[3] = v.w;
  }
  __syncthreads();

  const int wave = t >> 5;
  const int lane = t & 31;
  const int row  = lane & 15;
  const int koff = (lane >> 4) << 1;        // lanes 0-15: K+0/1, 16-31: K+2/3
  const int col  = (wave << 4) + (lane & 15);

  v8f acc = {};
  for (int k = 0; k < F; k += 4) {
    v2f a, b;
    a.x = lds[row * 132 + k + koff];
    a.y = lds[row * 132 + k + koff + 1];
    b.x = B[(size_t)(k + koff)     * F + col];
    b.y = B[(size_t)(k + koff + 1) * F + col];
    acc = __builtin_amdgcn_wmma_f32_16x16x4_f32(false, a, false, b,
                                                (short)0, acc, false, false);
  }
  // C/D layout: VGPR v -> M = v (lanes 0-15) or v+8 (lanes 16-31)
  const int mbase = m0 + ((lane < 16) ? 0 : 8);
#pragma unroll
  for (int v = 0; v < 8; ++v)
    C[(size_t)(mbase + v) * F + col] = acc[v];
}

// -------- per-node attention scalars: as=h.a_src, ad=h.a_dst (wave/node) ----
__global__ __launch_bounds__(256) void gat_alpha(
    const float* __restrict__ h, const float* __restrict__ a_src,
    const float* __restrict__ a_dst, float* __restrict__ as_,
    float* __restrict__ ad_) {
  const int node = (int)((blockIdx.x * (unsigned)blockDim.x + threadIdx.x) >> 5);
  const int lane = threadIdx.x & 31;
  if (node >= N_NODES) return;
  const float4 hv = *(const float4*)(h + (size_t)node * F + lane * 4);
  const float4 s4 = *(const float4*)(a_src + lane * 4);
  const float4 d4 = *(const float4*)(a_dst + lane * 4);
  float ps = hv.x * s4.x + hv.y * s4.y + hv.z * s4.z + hv.w * s4.w;
  float pd = hv.x * d4.x + hv.y * d4.y + hv.z * d4.z + hv.w * d4.w;
  for (int off = 16; off > 0; off >>= 1) {
    ps += __shfl_xor(ps, off, 32);
    pd += __shfl_xor(pd, off, 32);
  }
  if (lane == 0) { as_[node] = ps; ad_[node] = pd; }
}

// -------- fills --------
__global__ void fill_f32(float* p, float v, int n) {
  int i = blockIdx.x * blockDim.x + threadIdx.x;
  if (i < n) p[i] = v;
}
__global__ void fill_u32(unsigned* p, unsigned v, int n) {
  int i = blockIdx.x * blockDim.x + threadIdx.x;
  if (i < n) p[i] = v;
}

// -------- edge pass 1: e = leaky(as[s]+ad[d]); segment max over dst --------
__global__ void gat_edge1(const int* __restrict__ src, const int* __restrict__ dst,
                          const float* __restrict__ as_, const float* __restrict__ ad_,
                          float* __restrict__ ew, unsigned* __restrict__ mx) {
  int i = blockIdx.x * blockDim.x + threadIdx.x;
  if (i >= E2) return;
  int s, d;
  if (i < N_EDGES) { s = src[i]; d = dst[i]; } else { s = d = i - N_EDGES; }
  float e = as_[s] + ad_[d];
  e = e > 0.f ? e : 0.2f * e;
  ew[i] = e;
  atomicMax(&mx[d], enc_f(e));
}

// -------- edge pass 2: ex = exp(e - m[d]); z[d] += ex --------
__global__ void gat_edge2(const int* __restrict__ dst, float* __restrict__ ew,
                          const unsigned* __restrict__ mx, float* __restrict__ z) {
  int i = blockIdx.x * blockDim.x + threadIdx.x;
  if (i >= E2) return;
  int d = (i < N_EDGES) ? dst[i] : (i - N_EDGES);
  float ex = expf(ew[i] - dec_f(mx[d]));
  ew[i] = ex;
  atomicAdd(&z[d], ex);
}

// -------- edge pass 3: out[d] += h[s] * (ex / z[d])  (wave per edge) -------
__global__ __launch_bounds__(256) void gat_edge3(
    const int* __restrict__ src, const int* __restrict__ dst,
    const float* __restrict__ ew, const float* __restrict__ z,
    const float* __restrict__ h, float* __restrict__ out) {
  const int e    = (int)((blockIdx.x * (unsigned)blockDim.x + threadIdx.x) >> 5);
  const int lane = threadIdx.x & 31;
  if (e >= E2) return;
  int s, d;
  if (e < N_EDGES) { s = src[e]; d = dst[e]; } else { s = d = e - N_EDGES; }
  const float coef = ew[e] / z[d];
  const float4 hv = *(const float4*)(h + (size_t)s * F + lane * 4);
  float* o = out + (size_t)d * F + lane * 4;
  atomicAdd(o + 0, hv.x * coef);
  atomicAdd(o + 1, hv.y * coef);
  atomicAdd(o + 2, hv.z * coef);
  atomicAdd(o + 3, hv.w * coef);
}

// -------- out = selu(out + b) --------
__global__ void gat_bias_selu(float* __restrict__ out, const float* __restrict__ b,
                              int n) {
  int i = blockIdx.x * blockDim.x + threadIdx.x;
  if (i < n) out[i] = selu_f(out[i] + b[i & (F - 1)]);
}

// -------- pooling: pooled[batch[i]] += h[i]; cnt += 1 (wave per node) ------
__global__ __launch_bounds__(256) void gat_pool(
    const float* __restrict__ h, const int* __restrict__ batch,
    float* __restrict__ pooled, float* __restrict__ cnt) {
  const int node = (int)((blockIdx.x * (unsigned)blockDim.x + threadIdx.x) >> 5);
  const int lane = threadIdx.x & 31;
  if (node >= N_NODES) return;
  const int g = batch[node];
  const float4 hv = *(const float4*)(h + (size_t)node * F + lane * 4);
  float* p = pooled + (size_t)g * F + lane * 4;
  atomicAdd(p + 0, hv.x);
  atomicAdd(p + 1, hv.y);
  atomicAdd(p + 2, hv.z);
  atomicAdd(p + 3, hv.w);
  if (lane == 0) atomicAdd(&cnt[g], 1.f);
}

// -------- head: mean -> selu -> fc1 -> selu -> fc2; block per graph --------
__global__ __launch_bounds__(64) void gat_head(
    const float* __restrict__ pooled, const float* __restrict__ cnt,
    const float* __restrict__ fc1_w, const float* __restrict__ fc1_b,
    const float* __restrict__ fc2_w, const float* __restrict__ fc2_b,
    float* __restrict__ out) {
  __shared__ float sp[F];
  __shared__ float sf[NHID];
  const int g = blockIdx.x, t = threadIdx.x;
  float c = cnt[g]; c = c > 1.f ? c : 1.f;
  for (int i = t; i < F; i += NHID) sp[i] = selu_f(pooled[(size_t)g * F + i] / c);
  __syncthreads();
  float acc = fc1_b[t];
  for (int k = 0; k < F; ++k) acc += sp[k] * fc1_w[k * NHID + t];
  const float f = selu_f(acc);
  sf[t] = f;
  out[2 * NUM_GRAPHS + g * NHID + t] = f;   // feat after logits block
  __syncthreads();
  if (t < 2) {
    float a2 = fc2_b[t];
    for (int j = 0; j < NHID; ++j) a2 += sf[j] * fc2_w[j * 2 + t];
    out[g * 2 + t] = a2;                    // logits
  }
}

extern "C" void kernel_launch(void* const* d_in, const int* in_sizes, int n_in,
                              void* d_out, int out_size, void* d_ws, size_t ws_size,
                              hipStream_t stream) {
  const float* x      = (const float*)d_in[0];
  const int*   ei     = (const int*)d_in[1];
  const int*   batch  = (const int*)d_in[2];
  const float* W1     = (const float*)d_in[3];
  const float* asrc1  = (const float*)d_in[4];
  const float* adst1  = (const float*)d_in[5];
  const float* b1     = (const float*)d_in[6];
  const float* W2     = (const float*)d_in[7];
  const float* asrc2  = (const float*)d_in[8];
  const float* adst2  = (const float*)d_in[9];
  const float* b2     = (const float*)d_in[10];
  const float* fc1_w  = (const float*)d_in[11];
  const float* fc1_b  = (const float*)d_in[12];
  const float* fc2_w  = (const float*)d_in[13];
  const float* fc2_b  = (const float*)d_in[14];
  const int* src = ei;
  const int* dst = ei + N_EDGES;

  float* ws = (float*)d_ws;
  float*    bufA   = ws;                         // N*F  (h, pre-activation)
  float*    bufB   = bufA + (size_t)N_NODES * F; // N*F  (aggregated output)
  float*    ew     = bufB + (size_t)N_NODES * F; // E2
  float*    as_    = ew + E2;                    // N
  float*    ad_    = as_ + N_NODES;              // N
  float*    z_     = ad_ + N_NODES;              // N
  unsigned* mx     = (unsigned*)(z_ + N_NODES);  // N
  float*    pooled = (float*)(mx + N_NODES);     // 64*F
  float*    cnt    = pooled + NUM_GRAPHS * F;    // 64

  const int gGemm  = N_NODES / 16;               // 6250, exact
  const int gWaveN = (N_NODES * 32 + 255) / 256; // wave per node
  const int gEdge  = (E2 + 255) / 256;
  const int gWaveE = (int)(((size_t)E2 * 32 + 255) / 256);
  const int nNF    = N_NODES * F;

  for (int layer = 0; layer < 2; ++layer) {
    const float* hin  = (layer == 0) ? x : bufB;
    const float* W    = (layer == 0) ? W1 : W2;
    const float* asv  = (layer == 0) ? asrc1 : asrc2;
    const float* adv  = (layer == 0) ? adst1 : adst2;
    const float* bias = (layer == 0) ? b1 : b2;

    gat_gemm_f32<<<gGemm, 256, 0, stream>>>(hin, W, bufA);
    gat_alpha<<<gWaveN, 256, 0, stream>>>(bufA, asv, adv, as_, ad_);
    fill_u32<<<(N_NODES + 255) / 256, 256, 0, stream>>>(mx, 0u, N_NODES);
    fill_f32<<<(N_NODES + 255) / 256, 256, 0, stream>>>(z_, 0.f, N_NODES);
    fill_f32<<<(nNF + 255) / 256, 256, 0, stream>>>(bufB, 0.f, nNF);
    gat_edge1<<<gEdge, 256, 0, stream>>>(src, dst, as_, ad_, ew, mx);
    gat_edge2<<<gEdge, 256, 0, stream>>>(dst, ew, mx, z_);
    gat_edge3<<<gWaveE, 256, 0, stream>>>(src, dst, ew, z_, bufA, bufB);
    gat_bias_selu<<<(nNF + 255) / 256, 256, 0, stream>>>(bufB, bias, nNF);
  }

  fill_f32<<<(NUM_GRAPHS * F + NUM_GRAPHS + 255) / 256, 256, 0, stream>>>(
      pooled, 0.f, NUM_GRAPHS * F + NUM_GRAPHS);
  gat_pool<<<gWaveN, 256, 0, stream>>>(bufB, batch, pooled, cnt);
  gat_head<<<NUM_GRAPHS, NHID, 0, stream>>>(pooled, cnt, fc1_w, fc1_b,
                                            fc2_w, fc2_b, (float*)d_out);
}